// SAGE_22935125360937
// MI455X (gfx1250) — compile-verified
//
#include <hip/hip_runtime.h>

#define NNODES 50000
#define NEDGES 800000
#define DIM    128

typedef float v2f __attribute__((ext_vector_type(2)));
typedef float v8f __attribute__((ext_vector_type(8)));

// ---------------- CSR construction (done once, reused for 3 layers) --------

__global__ void zero_int2_kernel(int* __restrict__ a, int* __restrict__ b, int n) {
    int i = blockIdx.x * blockDim.x + threadIdx.x;
    if (i < n) { a[i] = 0; b[i] = 0; }
}

__global__ void count_deg_kernel(const long long* __restrict__ ei,
                                 int* __restrict__ deg, int nE) {
    int e = blockIdx.x * blockDim.x + threadIdx.x;
    if (e < nE) {
        int d = (int)ei[(size_t)nE + e];   // row 1 = dst
        atomicAdd(&deg[d], 1);
    }
}

// single-block exclusive scan over deg[] -> rowstart[] (N=50000, ~196 chunks)
__global__ void scan_kernel(const int* __restrict__ deg,
                            int* __restrict__ rowstart, int n) {
    __shared__ int smem[256];
    __shared__ int carry_s;
    if (threadIdx.x == 0) carry_s = 0;
    __syncthreads();
    for (int base = 0; base < n; base += 256) {
        int i = base + (int)threadIdx.x;
        int v = (i < n) ? deg[i] : 0;
        smem[threadIdx.x] = v;
        __syncthreads();
        for (int off = 1; off < 256; off <<= 1) {
            int t = (threadIdx.x >= (unsigned)off) ? smem[threadIdx.x - off] : 0;
            __syncthreads();
            smem[threadIdx.x] += t;
            __syncthreads();
        }
        if (i < n) rowstart[i] = carry_s + smem[threadIdx.x] - v;  // exclusive
        __syncthreads();
        if (threadIdx.x == 255) carry_s += smem[255];
        __syncthreads();
    }
    if (threadIdx.x == 0) rowstart[n] = carry_s;
}

__global__ void fill_csr_kernel(const long long* __restrict__ ei,
                                const int* __restrict__ rowstart,
                                int* __restrict__ cursor,
                                int* __restrict__ col, int nE) {
    int e = blockIdx.x * blockDim.x + threadIdx.x;
    if (e < nE) {
        int s = (int)ei[e];                // row 0 = src
        int d = (int)ei[(size_t)nE + e];   // row 1 = dst
        int pos = atomicAdd(&cursor[d], 1);
        col[rowstart[d] + pos] = s;
    }
}

// per-node insertion sort of neighbor lists -> CSR content (and therefore all
// fp summation orders) is bit-deterministic regardless of atomic scheduling.
__global__ void sort_csr_kernel(const int* __restrict__ rowstart,
                                int* __restrict__ col, int n) {
    int i = blockIdx.x * blockDim.x + threadIdx.x;
    if (i >= n) return;
    int s = rowstart[i], e = rowstart[i + 1];
    for (int a = s + 1; a < e; ++a) {
        int v = col[a];
        int b = a - 1;
        while (b >= s && col[b] > v) { col[b + 1] = col[b]; --b; }
        col[b + 1] = v;
    }
}

// ---------------- weight transpose: Wt[n][k] = W[k][n] (tiny, once) --------

__global__ void transpose_w_kernel(const float* __restrict__ W,
                                   float* __restrict__ Wt,
                                   int din, int dout) {
    int idx = blockIdx.x * blockDim.x + threadIdx.x;
    if (idx < din * dout) {
        int k = idx / dout;
        int c = idx - k * dout;           // coalesced read of W[k][c]
        Wt[(size_t)c * din + k] = W[idx];
    }
}

// ---------------- mean aggregation: one wave32 per node --------------------
// Batch-load up to 32 neighbor indices with one coalesced read, broadcast via
// shuffle; the 512B feature-row gathers then have no load->load dependency,
// so multiple stay in flight. Feature matrices (25.6MB) live in the 192MB L2.

__global__ void agg_mean_kernel(const float* __restrict__ h,
                                const int* __restrict__ rowstart,
                                const int* __restrict__ col,
                                float* __restrict__ agg, int n) {
    int wave = (int)((blockIdx.x * blockDim.x + threadIdx.x) >> 5);
    int lane = (int)(threadIdx.x & 31);
    if (wave >= n) return;
    int start = rowstart[wave];
    int end   = rowstart[wave + 1];
    float4 acc = make_float4(0.f, 0.f, 0.f, 0.f);
    for (int base = start; base < end; base += 32) {
        int rem = end - base;
        int cnt = rem < 32 ? rem : 32;
        int myIdx = (lane < cnt) ? col[base + lane] : 0;  // 1 coalesced load
        #pragma unroll 4
        for (int j = 0; j < cnt; ++j) {
            int s = __shfl(myIdx, j);                      // broadcast index
            float4 v = *(const float4*)(h + (size_t)s * DIM + lane * 4);
            acc.x += v.x; acc.y += v.y; acc.z += v.z; acc.w += v.w;
        }
    }
    int deg = end - start;
    float scale = 1.0f / (float)max(deg, 1);
    acc.x *= scale; acc.y *= scale; acc.z *= scale; acc.w *= scale;
    *(float4*)(agg + (size_t)wave * DIM + lane * 4) = acc;
}

// ---------------- fused dual-GEMM: out = agg@Wl + h@Wr + b (opt. ReLU) -----
// One wave per 16x16 output tile; K=128 stepped by 4 using
// V_WMMA_F32_16X16X4_F32 (exact fp32, matches reference numerics).
// A layout (16x4 f32): lane%16 = row; half-wave selects K pair {0,1}/{2,3}.
// B layout (4x16 f32): lane%16 = col; VGPR j + half-wave select K.
// Weights are pre-transposed so every operand load is a contiguous b64 with
// immediate offset (no per-load address arithmetic).
// C/D layout: VGPR i -> M = i + 8*(lane/16), N = lane%16.

__global__ void sage_gemm_kernel(const float* __restrict__ agg,
                                 const float* __restrict__ h,
                                 const float* __restrict__ Wlt,   // [dout][din]
                                 const float* __restrict__ Wrt,   // [dout][din]
                                 const float* __restrict__ bias,
                                 float* __restrict__ out,
                                 int nrows, int dout, int relu) {
    const int din = DIM;
    int waveId = (int)(threadIdx.x >> 5);
    int lane   = (int)(threadIdx.x & 31);
    int tilesPerRow = dout >> 4;
    int tile = (int)blockIdx.x * (int)(blockDim.x >> 5) + waveId;
    int totalTiles = (nrows >> 4) * tilesPerRow;
    if (tile >= totalTiles) return;      // wave-uniform: EXEC stays all-1s

    int mt = tile / tilesPerRow;
    int nt = tile - mt * tilesPerRow;
    int row0 = mt << 4, col0 = nt << 4;
    int r    = lane & 15;                // A row / B,C,D column within tile
    int half = lane >> 4;
    int colW = col0 + r;

    const float* aggRow = agg + (size_t)(row0 + r) * din + half * 2;
    const float* hRow   = h   + (size_t)(row0 + r) * din + half * 2;
    const float* wlRow  = Wlt + (size_t)colW * din + half * 2;
    const float* wrRow  = Wrt + (size_t)colW * din + half * 2;

    v8f c = {};
    #pragma unroll 8
    for (int k0 = 0; k0 < din; k0 += 4) {
        v2f a1 = *(const v2f*)(aggRow + k0);
        v2f a2 = *(const v2f*)(hRow + k0);
        v2f b1 = *(const v2f*)(wlRow + k0);
        v2f b2 = *(const v2f*)(wrRow + k0);
        c = __builtin_amdgcn_wmma_f32_16x16x4_f32(false, a1, false, b1,
                                                  (short)0, c, false, false);
        c = __builtin_amdgcn_wmma_f32_16x16x4_f32(false, a2, false, b2,
                                                  (short)0, c, false, false);
    }

    float bv = bias[colW];
    #pragma unroll
    for (int i = 0; i < 8; ++i) {
        int row = row0 + half * 8 + i;
        float val = c[i] + bv;
        if (relu) val = fmaxf(val, 0.0f);
        out[(size_t)row * dout + colW] = val;
    }
}

// ---------------------------------------------------------------------------

extern "C" void kernel_launch(void* const* d_in, const int* in_sizes, int n_in,
                              void* d_out, int out_size, void* d_ws, size_t ws_size,
                              hipStream_t stream) {
    const float*     x   = (const float*)d_in[0];
    const long long* ei  = (const long long*)d_in[1];  // int64 edge_index [2,E]
    const float*     Wl0 = (const float*)d_in[2];
    const float*     b0  = (const float*)d_in[3];
    const float*     Wr0 = (const float*)d_in[4];
    const float*     Wl1 = (const float*)d_in[5];
    const float*     b1  = (const float*)d_in[6];
    const float*     Wr1 = (const float*)d_in[7];
    const float*     Wl2 = (const float*)d_in[8];
    const float*     b2  = (const float*)d_in[9];
    const float*     Wr2 = (const float*)d_in[10];
    float* out = (float*)d_out;

    char* ws = (char*)d_ws;
    size_t off = 0;
    auto alloc = [&](size_t bytes) -> char* {
        char* p = ws + off;
        off += (bytes + 255) & ~(size_t)255;
        return p;
    };
    int*   deg      = (int*)alloc((size_t)NNODES * 4);
    int*   cursor   = (int*)alloc((size_t)NNODES * 4);
    int*   rowstart = (int*)alloc(((size_t)NNODES + 1) * 4);
    int*   col      = (int*)alloc((size_t)NEDGES * 4);
    float* Wl0t     = (float*)alloc((size_t)128 * 128 * 4);
    float* Wr0t     = (float*)alloc((size_t)128 * 128 * 4);
    float* Wl1t     = (float*)alloc((size_t)128 * 128 * 4);
    float* Wr1t     = (float*)alloc((size_t)128 * 128 * 4);
    float* Wl2t     = (float*)alloc((size_t)128 * 64 * 4);
    float* Wr2t     = (float*)alloc((size_t)128 * 64 * 4);
    float* aggBuf   = (float*)alloc((size_t)NNODES * DIM * 4);
    float* hA       = (float*)alloc((size_t)NNODES * DIM * 4);
    float* hB       = (float*)alloc((size_t)NNODES * DIM * 4);

    // CSR build (re-zeroed each call; sort makes content deterministic)
    zero_int2_kernel<<<(NNODES + 255) / 256, 256, 0, stream>>>(deg, cursor, NNODES);
    count_deg_kernel<<<(NEDGES + 255) / 256, 256, 0, stream>>>(ei, deg, NEDGES);
    scan_kernel<<<1, 256, 0, stream>>>(deg, rowstart, NNODES);
    fill_csr_kernel<<<(NEDGES + 255) / 256, 256, 0, stream>>>(ei, rowstart, cursor, col, NEDGES);
    sort_csr_kernel<<<(NNODES + 255) / 256, 256, 0, stream>>>(rowstart, col, NNODES);

    // weight transposes (tiny)
    transpose_w_kernel<<<(128 * 128 + 255) / 256, 256, 0, stream>>>(Wl0, Wl0t, 128, 128);
    transpose_w_kernel<<<(128 * 128 + 255) / 256, 256, 0, stream>>>(Wr0, Wr0t, 128, 128);
    transpose_w_kernel<<<(128 * 128 + 255) / 256, 256, 0, stream>>>(Wl1, Wl1t, 128, 128);
    transpose_w_kernel<<<(128 * 128 + 255) / 256, 256, 0, stream>>>(Wr1, Wr1t, 128, 128);
    transpose_w_kernel<<<(128 * 64 + 255) / 256, 256, 0, stream>>>(Wl2, Wl2t, 128, 64);
    transpose_w_kernel<<<(128 * 64 + 255) / 256, 256, 0, stream>>>(Wr2, Wr2t, 128, 64);

    int aggBlocks = (NNODES * 32 + 255) / 256;   // one wave32 per node
    int tiles128  = (NNODES / 16) * (128 / 16);  // 25000 tiles
    int tiles64   = (NNODES / 16) * (64 / 16);   // 12500 tiles

    // layer 0: h1 = relu(mean_agg(x)@Wl0 + x@Wr0 + b0)
    agg_mean_kernel<<<aggBlocks, 256, 0, stream>>>(x, rowstart, col, aggBuf, NNODES);
    sage_gemm_kernel<<<(tiles128 + 7) / 8, 256, 0, stream>>>(aggBuf, x, Wl0t, Wr0t, b0, hA,
                                                             NNODES, 128, 1);
    // layer 1: h2 = relu(mean_agg(h1)@Wl1 + h1@Wr1 + b1)
    agg_mean_kernel<<<aggBlocks, 256, 0, stream>>>(hA, rowstart, col, aggBuf, NNODES);
    sage_gemm_kernel<<<(tiles128 + 7) / 8, 256, 0, stream>>>(aggBuf, hA, Wl1t, Wr1t, b1, hB,
                                                             NNODES, 128, 1);
    // layer 2: out = mean_agg(h2)@Wl2 + h2@Wr2 + b2
    agg_mean_kernel<<<aggBlocks, 256, 0, stream>>>(hB, rowstart, col, aggBuf, NNODES);
    sage_gemm_kernel<<<(tiles64 + 7) / 8, 256, 0, stream>>>(aggBuf, hB, Wl2t, Wr2t, b2, out,
                                                            NNODES, 64, 0);
}